// SentenceEmbeddingV3_41300405518471
// MI455X (gfx1250) — compile-verified
//
#include <hip/hip_runtime.h>

// SentenceEmbeddingV3 GRU pipeline for gfx1250 (MI455X), all-fp32 via
// V_WMMA_F32_16X16X4_F32.
//   - Input projections: T-parallel WMMA GEMMs, 3 N-tiles per wave (A reuse),
//     writing gate-interleaved xW [T,B,H,4] (r,z,n,pad) so the scan loads all
//     three gates of a hidden unit with one b128. xW (<=134MB) stays resident
//     in the 192MB L2 between GEMM and scan.
//   - Recurrences: per-16-batch-row workgroups scanning 512 steps with
//     LDS-resident Whh and register-resident h state (D-fragment ownership is
//     time-invariant, so h never leaves the wave except the LDS mirror used to
//     re-read it in A-fragment layout).
// Workspace requirement: ~201 MB.

#define TT 512
#define BB 256

typedef __attribute__((ext_vector_type(2))) float v2f;
typedef __attribute__((ext_vector_type(4))) float v4f;
typedef __attribute__((ext_vector_type(8))) float v8f;

__device__ __forceinline__ v8f wmma_f32_k4(v2f a, v2f b, v8f c) {
  // D(16x16,f32) = A(16x4,f32) * B(4x16,f32) + C
  return __builtin_amdgcn_wmma_f32_16x16x4_f32(
      /*neg_a=*/false, a, /*neg_b=*/false, b,
      /*c_mod=*/(short)0, c, /*reuse_a=*/false, /*reuse_b=*/false);
}

__device__ __forceinline__ float sigm(float x) {
  return 1.0f / (1.0f + __expf(-x));
}
__device__ __forceinline__ float tanh_fast(float x) {
  // saturates cleanly at +-1 (no inf/inf NaN)
  return 2.0f / (1.0f + __expf(-2.0f * x)) - 1.0f;
}

// Gate-interleaved GEMM: logical out[M,3H] = in[M,K] * W[3H,K]^T + bias[3H],
// stored as out[m][n][gate] with stride 4 (pad lane), i.e. offset
// m*4H + n*4 + gate.  Each wave: 16 rows x 48 cols (3 N-tiles, shared A).
// block = 128 threads (4 waves), grid = (M/64, N/48), N = 3H.
__global__ void wmma_gemm_bias(const float* __restrict__ in,
                               const float* __restrict__ W,
                               const float* __restrict__ bias,
                               float* __restrict__ out, int M, int K, int N) {
  const int H = N / 3;
  const int lane = threadIdx.x & 31;
  const int wv = threadIdx.x >> 5;
  const int l16 = lane & 15;
  const int hi = lane >> 4;  // 0: K lo-pair, 1: K hi-pair
  const long m0 = ((long)blockIdx.x * 4 + wv) * 16;
  const int n0 = blockIdx.y * 48;

  // A lane layout: lanes 0-15 -> (M=lane, K=k0..k0+1), lanes 16-31 -> K=k0+2..3
  const float* ap = in + (m0 + l16) * (long)K + 2 * hi;
  // B mirrors A in K; col = lane&15.  B[k][c] = W[c][k].
  const float* bp = W + (long)(n0 + l16) * K + 2 * hi;

  v8f c[3];
#pragma unroll
  for (int j = 0; j < 3; j++) c[j] = (v8f){0.f, 0.f, 0.f, 0.f, 0.f, 0.f, 0.f, 0.f};

  for (int k0 = 0; k0 < K; k0 += 4) {
    const v2f a = *(const v2f*)(ap + k0);
#pragma unroll
    for (int j = 0; j < 3; j++) {
      const v2f b = *(const v2f*)(bp + (long)j * 16 * K + k0);
      c[j] = wmma_f32_k4(a, b, c[j]);
    }
  }

#pragma unroll
  for (int j = 0; j < 3; j++) {
    const int cbase = n0 + 16 * j;       // uniform; tile lies in one gate (16|H)
    const int gate = cbase / H;          // 0=r, 1=z, 2=n
    const int ncol = cbase - gate * H + l16;
    const float bv = bias[cbase + l16];
#pragma unroll
    for (int i = 0; i < 8; i++) {
      // D element i: row m0 + i + 8*hi
      out[(m0 + i + 8 * hi) * (long)(4 * H) + ncol * 4 + gate] = c[j][i] + bv;
    }
  }
}

// GRU recurrence scan. xw = precomputed input gates [T,B,H,4] (incl. bih).
// Workgroup handles 16 batch rows; H/16 waves, wave w owns hidden cols
// [16w,16w+16) of the r,z,n gate tiles (elementwise combine is wave-local and
// D-fragment aligned).
template <int H>
__global__ void gru_scan(const float* __restrict__ xw,
                         const float* __restrict__ Whh,   // [3H,H]
                         const float* __restrict__ bhh,   // [3H]
                         const float* __restrict__ h0,    // [B,H] or null
                         float* __restrict__ ys,          // [T,B,H] or null
                         float* __restrict__ hT, int hT_stride) {
  constexpr int NW = H / 16;
  constexpr int LSTR = H + 2;  // even pad: keeps b64 LDS loads 8B aligned
  __shared__ float whh_lds[3 * H * LSTR];
  __shared__ float h_lds[16 * LSTR];

  const int tid = threadIdx.x;
  const int lane = tid & 31;
  const int wv = tid >> 5;
  const int l16 = lane & 15;
  const int hi = lane >> 4;
  const int b0 = blockIdx.x * 16;

  // Stage Whh into LDS (padded rows).
  for (int idx = tid; idx < 3 * H * H; idx += NW * 32) {
    const int r = idx / H;
    const int cc = idx - r * H;
    whh_lds[r * LSTR + cc] = Whh[idx];
  }

  const int ncol = wv * 16 + l16;  // this lane's hidden column
  float hprev[8];
#pragma unroll
  for (int i = 0; i < 8; i++) {
    const int m = i + 8 * hi;
    const float v = h0 ? h0[(b0 + m) * H + ncol] : 0.0f;
    hprev[i] = v;
    h_lds[m * LSTR + ncol] = v;
  }
  const float bhr = bhh[ncol];
  const float bhz = bhh[H + ncol];
  const float bhn = bhh[2 * H + ncol];
  __syncthreads();

  const int rowA = l16 * LSTR + 2 * hi;
  const int rowR = (0 * H + wv * 16 + l16) * LSTR + 2 * hi;
  const int rowZ = (1 * H + wv * 16 + l16) * LSTR + 2 * hi;
  const int rowN = (2 * H + wv * 16 + l16) * LSTR + 2 * hi;

  for (int t = 0; t < TT; t++) {
    v8f dr = {0.f, 0.f, 0.f, 0.f, 0.f, 0.f, 0.f, 0.f};
    v8f dz = {0.f, 0.f, 0.f, 0.f, 0.f, 0.f, 0.f, 0.f};
    v8f dn = {0.f, 0.f, 0.f, 0.f, 0.f, 0.f, 0.f, 0.f};
#pragma unroll
    for (int ks = 0; ks < H; ks += 4) {
      const v2f a = *(const v2f*)&h_lds[rowA + ks];
      const v2f br = *(const v2f*)&whh_lds[rowR + ks];
      dr = wmma_f32_k4(a, br, dr);
      const v2f bz = *(const v2f*)&whh_lds[rowZ + ks];
      dz = wmma_f32_k4(a, bz, dz);
      const v2f bn = *(const v2f*)&whh_lds[rowN + ks];
      dn = wmma_f32_k4(a, bn, dn);
    }

    const float* xwrow = xw + ((size_t)t * BB + b0) * (4 * H);
    if (t + 1 < TT) {
      // warm L2/WGP$ for next step's gate quads (global_prefetch_b8)
      __builtin_prefetch(
          xwrow + (size_t)BB * (4 * H) + (size_t)(8 * hi) * (4 * H) + ncol * 4,
          0, 0);
    }

    float hnew[8];
#pragma unroll
    for (int i = 0; i < 8; i++) {
      const int m = i + 8 * hi;
      // one b128: (ir, iz, in, pad) for this lane's hidden unit
      const v4f g = *(const v4f*)(xwrow + (size_t)m * (4 * H) + ncol * 4);
      const float r = sigm(g.x + dr[i] + bhr);
      const float z = sigm(g.y + dz[i] + bhz);
      const float n = tanh_fast(g.z + r * (dn[i] + bhn));
      hnew[i] = (1.0f - z) * n + z * hprev[i];
    }

    __syncthreads();  // all waves done reading h_lds for this step
#pragma unroll
    for (int i = 0; i < 8; i++) {
      const int m = i + 8 * hi;
      h_lds[m * LSTR + ncol] = hnew[i];
      hprev[i] = hnew[i];
    }
    if (ys) {
#pragma unroll
      for (int i = 0; i < 8; i++) {
        ys[((size_t)t * BB + b0 + (i + 8 * hi)) * H + ncol] = hnew[i];
      }
    }
    __syncthreads();  // h_lds updated before next step's A loads
  }

#pragma unroll
  for (int i = 0; i < 8; i++) {
    hT[(b0 + i + 8 * hi) * hT_stride + ncol] = hprev[i];
  }
}

extern "C" void kernel_launch(void* const* d_in, const int* in_sizes, int n_in,
                              void* d_out, int out_size, void* d_ws,
                              size_t ws_size, hipStream_t stream) {
  // Input order: x, then p1(2 layers), p2(2), p3(3), each layer (Wih,Whh,bih,bhh)
  const float* x = (const float*)d_in[0];
  const int M = TT * BB;  // 131072 rows for all seq GEMMs

  float* xwb = (float*)d_ws;                      // [T,B,64,4] max (134MB)
  float* seqA = xwb + (size_t)M * 256;            // [T,B,64] max
  float* seqB = seqA + (size_t)M * 64;            // [T,B,64] max
  float* hT1 = seqB + (size_t)M * 64;             // [2,B,16] compress1 finals
  float* junk = hT1 + 2 * BB * 16;                // [B,64] discard sink
  float* out = (float*)d_out;                     // [B,128]

  auto gemm = [&](const float* in, int iw, int ib, float* o, int K, int N) {
    dim3 grid(M / 64, N / 48);
    wmma_gemm_bias<<<grid, 128, 0, stream>>>(
        in, (const float*)d_in[iw], (const float*)d_in[ib], o, M, K, N);
  };

  // ---- compress1 (p1), H=16, h0 = 0 ----
  gemm(x, 1, 3, xwb, 300, 48);
  gru_scan<16><<<BB / 16, 32, 0, stream>>>(
      xwb, (const float*)d_in[2], (const float*)d_in[4], nullptr, seqA, hT1, 16);
  gemm(seqA, 5, 7, xwb, 16, 48);
  gru_scan<16><<<BB / 16, 32, 0, stream>>>(
      xwb, (const float*)d_in[6], (const float*)d_in[8], nullptr, seqB,
      hT1 + BB * 16, 16);

  // ---- decode1 (p2), H=16, h0 = compress1 finals ----
  gemm(seqB, 9, 11, xwb, 16, 48);
  gru_scan<16><<<BB / 16, 32, 0, stream>>>(
      xwb, (const float*)d_in[10], (const float*)d_in[12], hT1, seqA, junk, 16);
  gemm(seqA, 13, 15, xwb, 16, 48);
  gru_scan<16><<<BB / 16, 32, 0, stream>>>(
      xwb, (const float*)d_in[14], (const float*)d_in[16], hT1 + BB * 16, seqB,
      junk, 16);

  // ---- decode2 (p3), H=64, h0 = 0; finals of layers 1,2 -> output concat ----
  gemm(seqB, 17, 19, xwb, 16, 192);
  gru_scan<64><<<BB / 16, 128, 0, stream>>>(
      xwb, (const float*)d_in[18], (const float*)d_in[20], nullptr, seqA, junk,
      64);
  gemm(seqA, 21, 23, xwb, 64, 192);
  gru_scan<64><<<BB / 16, 128, 0, stream>>>(
      xwb, (const float*)d_in[22], (const float*)d_in[24], nullptr, seqB,
      out + 64, 128);  // h3[-2] -> out[:,64:128]
  gemm(seqB, 25, 27, xwb, 64, 192);
  gru_scan<64><<<BB / 16, 128, 0, stream>>>(
      xwb, (const float*)d_in[26], (const float*)d_in[28], nullptr, nullptr,
      out, 128);       // h3[-1] -> out[:,0:64]
}